// BlockAttnRes_62328565400274
// MI455X (gfx1250) — compile-verified
//
#include <hip/hip_runtime.h>
#include <math.h>

// BlockAttnRes for MI455X (gfx1250, wave32).
// Phase 1: per-n logits via V_WMMA_F32_16X16X4_F32 (16 t-rows x D  times  D x [qa qm 0..]).
// Phase 2: softmax over n (8) + n-weighted sum of V, coalesced b128 streams, NT stores.

typedef __attribute__((ext_vector_type(2))) float v2f;
typedef __attribute__((ext_vector_type(4))) float v4f;
typedef __attribute__((ext_vector_type(8))) float v8f;

#define NBLK   7
#define NTOT   8      // blocks + partial
#define BB     2
#define TT     4096
#define DD     2048
#define TILE_T 16
#define EPSV   1e-6f

__device__ __forceinline__ const float* v_row(const float* __restrict__ blocks,
                                              const float* __restrict__ partial,
                                              int n, int b, int t) {
  if (n < NBLK)
    return blocks + (((size_t)n * BB + b) * TT + t) * (size_t)DD;
  return partial + (((size_t)b * TT + t)) * (size_t)DD;
}

__device__ __forceinline__ void fma4(v4f& acc, float w, const v4f x) {
  acc.x = fmaf(w, x.x, acc.x);
  acc.y = fmaf(w, x.y, acc.y);
  acc.z = fmaf(w, x.z, acc.z);
  acc.w = fmaf(w, x.w, acc.w);
}

// Fold norm weights into the pseudo-queries: q_eff = q * w  (dot(q, x_hat*w) == dot(q*w, x_hat))
__global__ void qprep_kernel(const float* __restrict__ qa, const float* __restrict__ qm,
                             const float* __restrict__ wa, const float* __restrict__ wm,
                             float* __restrict__ qeff) {
  int i = blockIdx.x * blockDim.x + threadIdx.x;
  if (i < DD) {
    qeff[i]      = qa[i] * wa[i];
    qeff[i + DD] = qm[i] * wm[i];
  }
}

__global__ __launch_bounds__(256)
void block_attn_res_kernel(const float* __restrict__ blocks,
                           const float* __restrict__ partial,
                           const float* __restrict__ qeff,   // [2*DD]: qa*wa, qm*wm
                           float* __restrict__ outA,
                           float* __restrict__ outM) {
  __shared__ float sDotA[NTOT][TILE_T];
  __shared__ float sDotM[NTOT][TILE_T];
  __shared__ float sS2  [NTOT][TILE_T];
  __shared__ float sAA  [NTOT][TILE_T];
  __shared__ float sAM  [NTOT][TILE_T];

  const int tid  = threadIdx.x;
  const int lane = tid & 31;
  const int wave = tid >> 5;                 // 0..7 == source block n
  const int tilesPerB = TT / TILE_T;
  const int b  = blockIdx.x / tilesPerB;
  const int t0 = (blockIdx.x % tilesPerB) * TILE_T;

  // ---------------- Phase 1: logits via WMMA, one wave per n ----------------
  {
    const int n     = wave;
    const int mrow  = lane & 15;             // A-matrix row M
    const int khalf = (lane >> 4) * 2;       // K pair {0,1} or {2,3}
    const float* arow = v_row(blocks, partial, n, b, t0 + mrow) + khalf;

    // B (4x16): col N=0 -> qa_eff, col N=1 -> qm_eff, cols 2..15 zero.
    const float* bp   = ((lane & 15) == 1) ? (qeff + DD) : qeff;
    const float bmask = ((lane & 15) < 2) ? 1.0f : 0.0f;

    v8f acc = {0.f, 0.f, 0.f, 0.f, 0.f, 0.f, 0.f, 0.f};
    float s2acc = 0.0f;

#pragma clang loop unroll_count(4)
    for (int d0 = 0; d0 < DD; d0 += 4) {
      v2f a  = *(const v2f*)(arow + d0);
      v2f bq = *(const v2f*)(bp + d0 + khalf);
      v2f bv;
      bv.x = bq.x * bmask;
      bv.y = bq.y * bmask;
      // D = A(16x4 f32) * B(4x16 f32) + C
      acc = __builtin_amdgcn_wmma_f32_16x16x4_f32(
          /*neg_a=*/false, a, /*neg_b=*/false, bv,
          /*c_mod=*/(short)0, acc, /*reuse_a=*/false, /*reuse_b=*/false);
      s2acc = fmaf(a.x, a.x, s2acc);
      s2acc = fmaf(a.y, a.y, s2acc);
    }

    // Sum-of-squares: lane l holds partial for row M=l&15 over its K half.
    float s2row = s2acc + __shfl_xor(s2acc, 16, 32);

    // D layout: element (M,N) -> VGPR M&7, lane N + (M>=8 ? 16 : 0).
    if (lane == 0)  { for (int v = 0; v < 8; ++v) sDotA[n][v]     = acc[v]; }
    if (lane == 16) { for (int v = 0; v < 8; ++v) sDotA[n][8 + v] = acc[v]; }
    if (lane == 1)  { for (int v = 0; v < 8; ++v) sDotM[n][v]     = acc[v]; }
    if (lane == 17) { for (int v = 0; v < 8; ++v) sDotM[n][8 + v] = acc[v]; }
    if (lane < 16)  { sS2[n][lane] = s2row; }
  }
  __syncthreads();

  // ---------------- Softmax over n for each (t, query) ----------------
  if (tid < 32) {
    const int qsel = tid >> 4;               // 0 = attn, 1 = mlp
    const int t    = tid & 15;
    const float invD     = 1.0f / (float)DD;
    const float invSqrtD = rsqrtf((float)DD);
    float lg[NTOT];
    for (int n = 0; n < NTOT; ++n) {
      float dot = qsel ? sDotM[n][t] : sDotA[n][t];
      float rms = rsqrtf(sS2[n][t] * invD + EPSV);
      lg[n] = dot * rms * invSqrtD;
    }
    float mx = lg[0];
    for (int n = 1; n < NTOT; ++n) mx = fmaxf(mx, lg[n]);
    float sum = 0.0f;
    for (int n = 0; n < NTOT; ++n) { lg[n] = __expf(lg[n] - mx); sum += lg[n]; }
    float inv = 1.0f / sum;
    for (int n = 0; n < NTOT; ++n) {
      if (qsel) sAM[n][t] = lg[n] * inv;
      else      sAA[n][t] = lg[n] * inv;
    }
  }
  __syncthreads();

  // ---------------- Phase 2: out[t,d] = sum_n a[n,t] * V[n,t,d] ----------------
  const int dbase = tid * 8;                 // 256 threads x 8 floats = 2048 = DD
  for (int t = 0; t < TILE_T; ++t) {
    v4f oa0 = {0,0,0,0}, oa1 = {0,0,0,0};
    v4f om0 = {0,0,0,0}, om1 = {0,0,0,0};
    for (int n = 0; n < NTOT; ++n) {
      const float wA = sAA[n][t];
      const float wM = sAM[n][t];
      const float* row = v_row(blocks, partial, n, b, t0 + t) + dbase;
      v4f x0 = *(const v4f*)(row);
      v4f x1 = *(const v4f*)(row + 4);
      fma4(oa0, wA, x0); fma4(oa1, wA, x1);
      fma4(om0, wM, x0); fma4(om1, wM, x1);
    }
    size_t o = (((size_t)b * TT) + t0 + t) * (size_t)DD + dbase;
    __builtin_nontemporal_store(oa0, (v4f*)(outA + o));
    __builtin_nontemporal_store(oa1, (v4f*)(outA + o + 4));
    __builtin_nontemporal_store(om0, (v4f*)(outM + o));
    __builtin_nontemporal_store(om1, (v4f*)(outM + o + 4));
  }
}

extern "C" void kernel_launch(void* const* d_in, const int* in_sizes, int n_in,
                              void* d_out, int out_size, void* d_ws, size_t ws_size,
                              hipStream_t stream) {
  const float* blocks  = (const float*)d_in[0];   // [7, 2, 4096, 2048]
  const float* partial = (const float*)d_in[1];   // [2, 4096, 2048]
  const float* qa      = (const float*)d_in[2];   // [2048]
  const float* qm      = (const float*)d_in[3];   // [2048]
  const float* wa      = (const float*)d_in[4];   // [2048]
  const float* wm      = (const float*)d_in[5];   // [2048]

  float* outA = (float*)d_out;                         // h_attn [2,4096,2048]
  float* outM = outA + (size_t)BB * TT * DD;           // h_mlp
  float* qeff = (float*)d_ws;                          // [2*2048]

  qprep_kernel<<<(DD + 255) / 256, 256, 0, stream>>>(qa, qm, wa, wm, qeff);

  const int nBlocks = BB * (TT / TILE_T);              // 512
  block_attn_res_kernel<<<nBlocks, 256, 0, stream>>>(blocks, partial, qeff, outA, outM);
}